// IntervalTimeEncoder_77653008712021
// MI455X (gfx1250) — compile-verified
//
#include <hip/hip_runtime.h>
#include <hip/hip_bf16.h>

// IntervalTimeEncoder for MI455X (gfx1250).
// out0[b][l][0:64] = W[clamp(floor((t[b][l+1]-t[b][l])/PER_TIME),0,511)] + bias
// out1[b][l]       = t[b][l]
// Store-bandwidth bound (134 MB out). CDNA5 path: async global->LDS staging of
// the 128KB embedding table (ASYNCcnt, GVS mode: one voffset VGPR + SGPR base,
// offset: immediate advances LDS dst and global src together) +
// ds_load_b128 gathers + NT b128 stores.

typedef float v4f __attribute__((ext_vector_type(4)));

#define B_DIM    64
#define L_DIM    8192
#define TDIM     64
#define NBUCKET  512
#define PER_TIME_F 1953.125f     // 1e6 / 512, exactly representable in f32
#define ROWS_PER_BLOCK 1024
#define THREADS  256
#define STAGE_STEP (THREADS * 16)                       // 4096 B per round
#define STAGE_ITERS (NBUCKET * TDIM * 4 / STAGE_STEP)   // 32 rounds = 128 KB

// Issue all async copies with compile-time 'offset:' immediates (added by HW
// to BOTH the LDS destination and the global source address, ISA §10 async).
template <int I, int N>
__device__ __forceinline__ void stage_async(unsigned ldsa, unsigned voff,
                                            const float* gbase) {
    if constexpr (I < N) {
        asm volatile("global_load_async_to_lds_b128 %0, %1, %2 offset:%c3"
                     :: "v"(ldsa), "v"(voff), "s"(gbase), "i"(I * STAGE_STEP)
                     : "memory");
        stage_async<I + 1, N>(ldsa, voff, gbase);
    }
}

__global__ __launch_bounds__(THREADS)
void IntervalTimeEncoder_kernel(const float* __restrict__ ts,
                                const float* __restrict__ W,
                                const float* __restrict__ bias,
                                float* __restrict__ out0,
                                float* __restrict__ out1) {
    __shared__ float sW[NBUCKET * TDIM];   // 128 KB: whole embedding table

    const int tid = threadIdx.x;

    // ---- Stage W into LDS with CDNA5 async global->LDS B128 copies ------
    // Low 32 bits of the generic shared pointer == wave-relative LDS byte
    // offset (LDS aperture mapping, ISA §10.2).
    {
        const unsigned lds_base =
            (unsigned)(unsigned long long)(uintptr_t)(&sW[0]);
        const unsigned ldsa = lds_base + (unsigned)(tid * 16);
        const unsigned voff = (unsigned)(tid * 16);
        stage_async<0, STAGE_ITERS>(ldsa, voff, W);
    }

    // Overlap bias load + address setup with the in-flight async copies.
    const int rsub  = tid >> 4;          // row within 16-row tile
    const int dlane = (tid & 15) << 2;   // float offset within embedding row
    const v4f bb = *(const v4f*)(bias + dlane);

    const long long r0 = (long long)blockIdx.x * ROWS_PER_BLOCK;
    const int b  = (int)(r0 >> 13);          // ROWS_PER_BLOCK divides L_DIM
    const int l0 = (int)(r0 & (L_DIM - 1));

    const float* tp = ts + (long long)b * (L_DIM + 1) + l0 + rsub;
    float*       o0 = out0 + (r0 + rsub) * TDIM + dlane;
    float*       o1 = out1 + r0 + rsub;

    asm volatile("s_wait_asynccnt 0" ::: "memory");
    __syncthreads();

    // ---- Gather + stream out -------------------------------------------
#pragma unroll 4
    for (int it = 0; it < ROWS_PER_BLOCK / 16; ++it) {
        const float t0 = tp[0];
        const float t1 = tp[1];
        const float diff = t1 - t0;
        int idx = (int)floorf(diff / PER_TIME_F);
        idx = idx < 0 ? 0 : idx;                       // relu on index
        idx = idx > (NBUCKET - 1) ? (NBUCKET - 1) : idx;

        const v4f wv = *(const v4f*)(&sW[idx * TDIM + dlane]); // ds_load_b128
        const v4f ov = wv + bb;

        __builtin_nontemporal_store(ov, (v4f*)o0);
        if (dlane == 0) {
            __builtin_nontemporal_store(t0, o1);
        }
        tp += 16;
        o0 += 16 * TDIM;
        o1 += 16;
    }
}

extern "C" void kernel_launch(void* const* d_in, const int* in_sizes, int n_in,
                              void* d_out, int out_size, void* d_ws, size_t ws_size,
                              hipStream_t stream) {
    (void)in_sizes; (void)n_in; (void)out_size; (void)d_ws; (void)ws_size;
    // setup_inputs order: input(int32, unused), timestamp, W, b
    const float* ts   = (const float*)d_in[1];
    const float* W    = (const float*)d_in[2];
    const float* bias = (const float*)d_in[3];

    float* out0 = (float*)d_out;                                   // [B,L,64]
    float* out1 = out0 + (size_t)B_DIM * L_DIM * TDIM;             // [B,L]

    const int rows = B_DIM * L_DIM;                 // 524288
    dim3 grid(rows / ROWS_PER_BLOCK);               // 512 blocks
    dim3 block(THREADS);                            // 8 wave32s
    IntervalTimeEncoder_kernel<<<grid, block, 0, stream>>>(ts, W, bias, out0, out1);
}